// PortfolioGenerator_35064113004829
// MI455X (gfx1250) — compile-verified
//
#include <hip/hip_runtime.h>
#include <stdint.h>

#define ROWS    4096
#define NCOLS   8192
#define GTOP    20
#define THREADS 256
#define CHUNK   32          // elements per thread held in registers

// One compare-exchange level with constant stride J inside a thread's chunk.
// i/j are compile-time after unrolling, so `r` stays in VGPRs.
template<int J>
__device__ __forceinline__ void local_level(unsigned long long* r,
                                            unsigned base, unsigned k) {
  #pragma unroll
  for (int p = 0; p < CHUNK / 2; ++p) {
    const int i = ((p & ~(J - 1)) << 1) | (p & (J - 1));
    const int j = i | J;
    const bool up = (((base + (unsigned)i) & k) == 0u);
    unsigned long long a = r[i], b = r[j];
    if ((a > b) == up) { r[i] = b; r[j] = a; }
  }
}

// Full bitonic network for stages k=2..32 (all intra-chunk).
__device__ __forceinline__ void local_sort32(unsigned long long* r, unsigned base) {
  local_level<1>(r, base, 2);
  local_level<2>(r, base, 4);  local_level<1>(r, base, 4);
  local_level<4>(r, base, 8);  local_level<2>(r, base, 8);  local_level<1>(r, base, 8);
  local_level<8>(r, base, 16); local_level<4>(r, base, 16);
  local_level<2>(r, base, 16); local_level<1>(r, base, 16);
  local_level<16>(r, base, 32); local_level<8>(r, base, 32);
  local_level<4>(r, base, 32);  local_level<2>(r, base, 32);
  local_level<1>(r, base, 32);
}

// j=16..1 tail of a merge stage k>=64 (intra-chunk).
__device__ __forceinline__ void local_tail(unsigned long long* r,
                                           unsigned base, unsigned k) {
  local_level<16>(r, base, k); local_level<8>(r, base, k);
  local_level<4>(r, base, k);  local_level<2>(r, base, k);
  local_level<1>(r, base, k);
}

// One workgroup per row. LDS layout (dynamic, 96 KB):
//   [0, 64K)   : u64 sort keys (key<<32 | index)
//   [64K, 96K) : f32 staging for async row copy; reused as softmax scratch
__global__ __launch_bounds__(THREADS)
void portfolio_sort_kernel(const float* __restrict__ in,
                           float* __restrict__ wout,
                           int* __restrict__ iout) {
  extern __shared__ char smem[];
  unsigned long long* keys = (unsigned long long*)smem;              // 64 KB
  float* stage = (float*)(smem + (size_t)NCOLS * 8);                 // 32 KB
  const int tid = threadIdx.x;
  const int row = blockIdx.x;
  const float* in_row = in + (size_t)row * NCOLS;
  const unsigned base = (unsigned)tid * CHUNK;

  // ---- CDNA5 async global->LDS copy of the row into the staging region ----
  {
    const unsigned stage_base = (unsigned)(NCOLS * 8);  // LDS byte offset
    for (int c = tid; c < NCOLS / 4; c += THREADS) {    // 2048 x 16B chunks
      unsigned loff = stage_base + (unsigned)c * 16u;
      const float* gptr = in_row + c * 4;
      asm volatile("global_load_async_to_lds_b128 %0, %1, off"
                   :: "v"(loff), "v"(gptr)
                   : "memory");
    }
    asm volatile("s_wait_asynccnt 0" ::: "memory");
  }
  __syncthreads();

  unsigned long long r[CHUNK];

  // ---- pack this thread's 32 elements into sortable u64 keys and run the
  //      k=2..32 bitonic stages entirely in registers ----
  #pragma unroll
  for (int t = 0; t < CHUNK; ++t) {
    unsigned i = base + (unsigned)t;
    unsigned b = __float_as_uint(stage[i]);
    unsigned mapped = (b & 0x80000000u) ? ~b : (b | 0x80000000u); // asc f32
    unsigned key = ~mapped;                                       // desc f32
    r[t] = ((unsigned long long)key << 32) | i;
  }
  local_sort32(r, base);
  #pragma unroll
  for (int t = 0; t < CHUNK; ++t) keys[base + t] = r[t];
  __syncthreads();

  // ---- merge stages k=64..8192: LDS passes for j>=32, register tail j<=16 ----
  for (unsigned k = 64; k <= (unsigned)NCOLS; k <<= 1) {
    for (unsigned j = k >> 1; j >= CHUNK; j >>= 1) {
      for (unsigned p = tid; p < NCOLS / 2; p += THREADS) {
        unsigned i   = ((p & ~(j - 1u)) << 1) | (p & (j - 1u));
        unsigned ixj = i | j;
        unsigned long long a = keys[i];
        unsigned long long b = keys[ixj];
        bool up = ((i & k) == 0u);
        if ((a > b) == up) { keys[i] = b; keys[ixj] = a; }
      }
      __syncthreads();
    }
    #pragma unroll
    for (int t = 0; t < CHUNK; ++t) r[t] = keys[base + t];
    local_tail(r, base, k);
    #pragma unroll
    for (int t = 0; t < CHUNK; ++t) keys[base + t] = r[t];
    __syncthreads();
  }

  // ---- gather top-G / bottom-G scores into scratch ----
  if (tid < 2 * GTOP) {
    int s_r = (tid < GTOP) ? tid : (NCOLS - GTOP + (tid - GTOP));
    unsigned idx = (unsigned)keys[s_r];
    float s = in_row[idx];
    stage[tid] = (tid < GTOP) ? s : -s;   // bottom group uses negated scores
  }
  __syncthreads();

  // ---- two tiny softmaxes (20 elements each), one lane per group ----
  if (tid < 2) {
    float* v = stage + tid * GTOP;
    float m = v[0];
    for (int t = 1; t < GTOP; ++t) m = fmaxf(m, v[t]);
    float sum = 0.0f;
    for (int t = 0; t < GTOP; ++t) { float e = __expf(v[t] - m); v[t] = e; sum += e; }
    float inv = 1.0f / sum;
    for (int t = 0; t < GTOP; ++t) v[t] *= inv;
  }
  __syncthreads();

  // ---- outputs: sorted indices (coalesced) + weight scatter (permutation,
  //      every column written exactly once; no zero pass needed) ----
  float* wrow = wout + (size_t)row * NCOLS;
  int*   irow = iout + (size_t)row * NCOLS;
  for (int q = tid; q < NCOLS; q += THREADS) {
    unsigned long long kk = keys[q];
    unsigned idx = (unsigned)kk;
    irow[q] = (int)idx;
    float w = 0.0f;
    if (q < GTOP)                 w =  stage[q];
    else if (q >= NCOLS - GTOP)   w = -stage[GTOP + (q - (NCOLS - GTOP))];
    wrow[idx] = w;
  }
}

extern "C" void kernel_launch(void* const* d_in, const int* in_sizes, int n_in,
                              void* d_out, int out_size, void* d_ws, size_t ws_size,
                              hipStream_t stream) {
  (void)in_sizes; (void)n_in; (void)out_size; (void)d_ws; (void)ws_size;
  const float* in = (const float*)d_in[0];
  float* wout = (float*)d_out;                                   // weights first
  int*   iout = (int*)((float*)d_out + (size_t)ROWS * NCOLS);    // then indices
  size_t shmem = (size_t)NCOLS * 8 + (size_t)NCOLS * 4;          // 96 KB
  hipLaunchKernelGGL(portfolio_sort_kernel, dim3(ROWS), dim3(THREADS),
                     shmem, stream, in, wout, iout);
}